// EpidemicDynamics_23270132810378
// MI455X (gfx1250) — compile-verified
//
#include <hip/hip_runtime.h>

// out = 0.1 * x * (A @ (1 - x))  — bandwidth-bound matvec (1 GiB of A @ 23.3 TB/s ≈ 46 us floor).
// MAC engine: V_WMMA_F32_16X16X4_F32 (exact fp32), one 16-row block per wave, split-K partials,
// deterministic fixed-order reduction in a finalize kernel.

typedef __attribute__((ext_vector_type(2))) float v2f;
typedef __attribute__((ext_vector_type(4))) float v4f;
typedef __attribute__((ext_vector_type(8))) float v8f;

#define BLOCK 256          // 8 waves (wave32)
#define ROWS_PER_WG 128    // 8 waves * 16 rows each

// partial[chunk*n + row] = sum over this K-chunk of A[row][k] * (1 - x[k])
__global__ __launch_bounds__(BLOCK)
void matvec_wmma_f32(const float* __restrict__ A, const float* __restrict__ x,
                     float* __restrict__ partial, int n, int kc) {
    extern __shared__ float sv[];            // kc floats of (1 - x) for this K-chunk
    const int tid   = threadIdx.x;
    const int kBase = blockIdx.y * kc;

    // Stage (1 - x) slice into LDS, coalesced b128 loads / ds_store_b128.
    for (int idx = tid * 4; idx < kc; idx += BLOCK * 4) {
        v4f r = *(const v4f*)(x + kBase + idx);
        v4f m; m.x = 1.0f - r.x; m.y = 1.0f - r.y; m.z = 1.0f - r.z; m.w = 1.0f - r.w;
        *(v4f*)(sv + idx) = m;
    }
    __syncthreads();

    const int wave    = tid >> 5;
    const int lane    = tid & 31;
    const int half    = lane >> 4;           // 0: K=k,k+1   1: K=k+2,k+3
    const int mrow    = lane & 15;           // matrix row within 16-row tile
    const int rowBase = blockIdx.x * ROWS_PER_WG + wave * 16;

    // A-fragment source: lane reads 2 consecutive f32 of its row (global_load_b64),
    // exactly the 16x4 f32 A-frag VGPR layout. Every byte of A is consumed once.
    const float* ap = A + (size_t)(rowBase + mrow) * n + (size_t)kBase + half * 2;
    // B-fragment source: (1-x) broadcast across all 16 columns, from LDS (ds_load_b64).
    const float* bp = sv + half * 2;

    v8f acc0 = {};
    v8f acc1 = {};
    #pragma unroll 4
    for (int k = 0; k < kc; k += 8) {
        v2f a0 = *(const v2f*)(ap + k);
        v2f b0 = *(const v2f*)(bp + k);
        acc0 = __builtin_amdgcn_wmma_f32_16x16x4_f32(false, a0, false, b0,
                                                     (short)0, acc0, false, false);
        v2f a1 = *(const v2f*)(ap + k + 4);
        v2f b1 = *(const v2f*)(bp + k + 4);
        acc1 = __builtin_amdgcn_wmma_f32_16x16x4_f32(false, a1, false, b1,
                                                     (short)0, acc1, false, false);
    }

    // All 16 columns of D are identical. D layout: VGPR r, lanes 0-15 -> M=r,
    // lanes 16-31 -> M=8+r. Lane 0 writes rows rowBase+0..7, lane 16 rows rowBase+8..15.
    if ((lane & 15) == 0) {
        v8f acc;
        #pragma unroll
        for (int r = 0; r < 8; ++r) acc[r] = acc0[r] + acc1[r];
        size_t off = (size_t)blockIdx.y * (size_t)n + (size_t)(rowBase + half * 8);
        v4f lo = {acc[0], acc[1], acc[2], acc[3]};
        v4f hi = {acc[4], acc[5], acc[6], acc[7]};
        *(v4f*)(partial + off)     = lo;
        *(v4f*)(partial + off + 4) = hi;
    }
}

// out[i] = 0.1 * x[i] * sum_c partial[c*n + i]   (fixed summation order -> deterministic)
__global__ __launch_bounds__(BLOCK)
void finalize_k(const float* __restrict__ x, const float* __restrict__ partial,
                float* __restrict__ out, int n, int nchunk) {
    int i = blockIdx.x * blockDim.x + threadIdx.x;
    if (i < n) {
        float s = 0.0f;
        for (int c = 0; c < nchunk; ++c) s += partial[(size_t)c * n + i];
        out[i] = 0.1f * x[i] * s;
    }
}

extern "C" void kernel_launch(void* const* d_in, const int* in_sizes, int n_in,
                              void* d_out, int out_size, void* d_ws, size_t ws_size,
                              hipStream_t stream) {
    // setup_inputs order: t (unused), x [n], A [n*n]
    const float* x = (const float*)d_in[1];
    const float* A = (const float*)d_in[2];
    float* out     = (float*)d_out;
    const int n    = in_sizes[1];            // 16384

    // Split-K factor: 8 chunks -> 1024 WGs / 8192 waves (enough to saturate HBM),
    // shrink if workspace is smaller than nchunk*n floats.
    int nchunk = 8;
    while (nchunk > 1 && (size_t)nchunk * (size_t)n * sizeof(float) > ws_size) nchunk >>= 1;
    const int kc = n / nchunk;

    float* partial = (float*)d_ws;
    dim3 grid(n / ROWS_PER_WG, nchunk);
    size_t lds_bytes = (size_t)kc * sizeof(float);
    hipLaunchKernelGGL(matvec_wmma_f32, grid, dim3(BLOCK), lds_bytes, stream,
                       A, x, partial, n, kc);

    hipLaunchKernelGGL(finalize_k, dim3((n + BLOCK - 1) / BLOCK), dim3(BLOCK), 0, stream,
                       x, partial, out, n, nchunk);
}